// VisionTransformerBlock_25572235280550
// MI455X (gfx1250) — compile-verified
//
#include <hip/hip_runtime.h>
#include <math.h>

// ---------------------------------------------------------------------------
// VisionTransformerBlock for MI455X (gfx1250, wave32, WMMA).
//
// Rewrite: no softmax between the attention matmuls, so
//   (Q K^T) V == Q (K^T V);  K^T V is 64x64 per head (51.6 GFLOP -> 3.2 GFLOP,
//   and no 1024x1024 score matrices).
// All GEMMs use V_WMMA_F32_16X16X4_F32 (full fp32 matrix pipe).
// Big GEMMs: 256x128 block tile, 8 waves x (64x64), software-pipelined
// float4 global loads -> LDS slabs.
// ---------------------------------------------------------------------------

#define B_      16
#define P_      1024
#define E_      768
#define HID_    768
#define HEADS_  12
#define DH_     64
#define QKV_N   (3 * HID_)      // 2304
#define NPE     (P_ * E_)       // 786432 elements per batch sample

typedef __attribute__((ext_vector_type(2))) float v2f;
typedef __attribute__((ext_vector_type(8))) float v8f;

__device__ __forceinline__ v8f wmma4(v2f a, v2f b, v8f c) {
    return __builtin_amdgcn_wmma_f32_16x16x4_f32(
        /*neg_a=*/false, a, /*neg_b=*/false, b,
        /*c_mod=*/(short)0, c, /*reuse_a=*/false, /*reuse_b=*/false);
}

__device__ __forceinline__ v8f v8f_zero() {
    v8f z = {0.f, 0.f, 0.f, 0.f, 0.f, 0.f, 0.f, 0.f};
    return z;
}

// ---------------------------------------------------------------------------
// Kernel 1: per-batch mean / rstd over the full (P,E) slice.
// ---------------------------------------------------------------------------
__global__ __launch_bounds__(256) void stats_kernel(const float* __restrict__ x,
                                                    float* __restrict__ stats) {
    const int b = blockIdx.x;
    const int t = threadIdx.x;
    const float* xb = x + (size_t)b * NPE;
    double s = 0.0, s2 = 0.0;
    for (int i = t; i < NPE; i += 256) {
        float v = xb[i];
        s  += (double)v;
        s2 += (double)v * (double)v;
    }
    __shared__ double sh[256], sh2[256];
    sh[t] = s; sh2[t] = s2;
    __syncthreads();
    for (int off = 128; off > 0; off >>= 1) {
        if (t < off) { sh[t] += sh[t + off]; sh2[t] += sh2[t + off]; }
        __syncthreads();
    }
    if (t == 0) {
        double mean = sh[0] / (double)NPE;
        double var  = sh2[0] / (double)NPE - mean * mean;
        stats[2 * b]     = (float)mean;
        stats[2 * b + 1] = (float)(1.0 / sqrt(var + 1e-5));
    }
}

// ---------------------------------------------------------------------------
// Kernel 2: fused LayerNorm + QKV GEMM (16384x768 @ 768x2304).
// Block tile 256x128, 8 waves, each wave a 64x64 tile (16 accumulators).
// Software-pipelined: prefetch next K-slab into regs during WMMA stage.
// Writes q/k/v directly in [B][H][P][DH] layout.
// ---------------------------------------------------------------------------
__global__ __launch_bounds__(256) void qkv_kernel(
    const float* __restrict__ x, const float* __restrict__ lnw,
    const float* __restrict__ lnb, const float* __restrict__ qkv,
    const float* __restrict__ stats,
    float* __restrict__ qb, float* __restrict__ kb, float* __restrict__ vb) {
    __shared__ float As[256][20];   // row base 80B (16B aligned); 20m%64 distinct
    __shared__ float Bs[16][132];   // row base 528B (16B aligned)

    const int t    = threadIdx.x;
    const int lane = t & 31;
    const int lr   = lane & 15;
    const int lh   = lane >> 4;
    const int w    = t >> 5;
    const int wm   = w >> 1, wn = w & 1;     // 4x2 wave grid
    const int n0   = blockIdx.x * 128;
    const int m0   = blockIdx.y * 256;

    v8f acc[4][4];
#pragma unroll
    for (int mi = 0; mi < 4; ++mi)
#pragma unroll
        for (int ni = 0; ni < 4; ++ni) acc[mi][ni] = v8f_zero();

    float4 aReg[4], bReg[2];

    // prefetch slab k0 = 0 (LayerNorm fused into the A load)
#pragma unroll
    for (int i = 0; i < 4; ++i) {
        int lin = i * 256 + t;              // 1024 float4s = 256x16 slab
        int row = lin >> 2, c4 = (lin & 3) * 4;
        int gm  = m0 + row;
        int bb  = gm >> 10, p = gm & 1023;
        float mean = stats[2 * bb], rstd = stats[2 * bb + 1];
        float4 xv = *(const float4*)(x   + (size_t)gm * E_ + c4);
        float4 wv = *(const float4*)(lnw + (size_t)p  * E_ + c4);
        float4 bv = *(const float4*)(lnb + (size_t)p  * E_ + c4);
        aReg[i].x = (xv.x - mean) * rstd * wv.x + bv.x;
        aReg[i].y = (xv.y - mean) * rstd * wv.y + bv.y;
        aReg[i].z = (xv.z - mean) * rstd * wv.z + bv.z;
        aReg[i].w = (xv.w - mean) * rstd * wv.w + bv.w;
    }
#pragma unroll
    for (int i = 0; i < 2; ++i) {
        int lin = i * 256 + t;              // 512 float4s = 16x128 slab
        int row = lin >> 5, c4 = (lin & 31) * 4;
        bReg[i] = *(const float4*)(qkv + (size_t)row * QKV_N + n0 + c4);
    }

    for (int k0 = 0; k0 < E_; k0 += 16) {
        // commit prefetched slab to LDS
#pragma unroll
        for (int i = 0; i < 4; ++i) {
            int lin = i * 256 + t;
            int row = lin >> 2, c4 = (lin & 3) * 4;
            *(float4*)&As[row][c4] = aReg[i];
        }
#pragma unroll
        for (int i = 0; i < 2; ++i) {
            int lin = i * 256 + t;
            int row = lin >> 5, c4 = (lin & 31) * 4;
            *(float4*)&Bs[row][c4] = bReg[i];
        }
        __syncthreads();

        // prefetch next slab (overlaps the WMMA stage below)
        if (k0 + 16 < E_) {
            const int kn = k0 + 16;
#pragma unroll
            for (int i = 0; i < 4; ++i) {
                int lin = i * 256 + t;
                int row = lin >> 2, c4 = (lin & 3) * 4;
                int gm  = m0 + row;
                int bb  = gm >> 10, p = gm & 1023;
                int e   = kn + c4;
                float mean = stats[2 * bb], rstd = stats[2 * bb + 1];
                float4 xv = *(const float4*)(x   + (size_t)gm * E_ + e);
                float4 wv = *(const float4*)(lnw + (size_t)p  * E_ + e);
                float4 bv = *(const float4*)(lnb + (size_t)p  * E_ + e);
                aReg[i].x = (xv.x - mean) * rstd * wv.x + bv.x;
                aReg[i].y = (xv.y - mean) * rstd * wv.y + bv.y;
                aReg[i].z = (xv.z - mean) * rstd * wv.z + bv.z;
                aReg[i].w = (xv.w - mean) * rstd * wv.w + bv.w;
            }
#pragma unroll
            for (int i = 0; i < 2; ++i) {
                int lin = i * 256 + t;
                int row = lin >> 5, c4 = (lin & 31) * 4;
                bReg[i] = *(const float4*)(qkv + (size_t)(kn + row) * QKV_N + n0 + c4);
            }
        }

        // WMMA stage: 4 k-steps x 16 tiles
#pragma unroll
        for (int ks = 0; ks < 4; ++ks) {
            const int kk = ks * 4 + 2 * lh;
            v2f af[4], bf[4];
#pragma unroll
            for (int mi = 0; mi < 4; ++mi) {
                int m = wm * 64 + mi * 16 + lr;
                af[mi].x = As[m][kk];
                af[mi].y = As[m][kk + 1];
            }
#pragma unroll
            for (int ni = 0; ni < 4; ++ni) {
                int n = wn * 64 + ni * 16 + lr;
                bf[ni].x = Bs[kk][n];
                bf[ni].y = Bs[kk + 1][n];
            }
#pragma unroll
            for (int mi = 0; mi < 4; ++mi)
#pragma unroll
                for (int ni = 0; ni < 4; ++ni)
                    acc[mi][ni] = wmma4(af[mi], bf[ni], acc[mi][ni]);
        }
        __syncthreads();
    }

    // Epilogue: scatter into q/k/v [B][H][P][DH]
#pragma unroll
    for (int mi = 0; mi < 4; ++mi) {
#pragma unroll
        for (int ni = 0; ni < 4; ++ni) {
            int n = n0 + wn * 64 + ni * 16 + lr;
            int h = n / 192, r = n % 192;
            float* dst = (r < 64) ? qb : (r < 128) ? kb : vb;
            int d = r & 63;
#pragma unroll
            for (int v = 0; v < 8; ++v) {
                int gm = m0 + wm * 64 + mi * 16 + v + 8 * lh;
                int bb = gm >> 10, p = gm & 1023;
                dst[(((size_t)bb * HEADS_ + h) * P_ + p) * DH_ + d] = acc[mi][ni][v];
            }
        }
    }
}

// ---------------------------------------------------------------------------
// Kernel 3: M = K^T V per (b,h): 64x64 = (64x1024)@(1024x64).
// 8 waves split the K=1024 reduction; cross-wave reduce in LDS.
// Every element is read exactly once -> direct global loads are optimal.
// ---------------------------------------------------------------------------
__global__ __launch_bounds__(256) void ktv_kernel(const float* __restrict__ kb,
                                                  const float* __restrict__ vb,
                                                  float* __restrict__ Mb) {
    const int bh   = blockIdx.x;
    const int t    = threadIdx.x;
    const int lane = t & 31, lr = lane & 15, lh = lane >> 4;
    const int w    = t >> 5;
    const float* Kp = kb + (size_t)bh * P_ * DH_;
    const float* Vp = vb + (size_t)bh * P_ * DH_;

    v8f acc[4][4];
#pragma unroll
    for (int mi = 0; mi < 4; ++mi)
#pragma unroll
        for (int ni = 0; ni < 4; ++ni) acc[mi][ni] = v8f_zero();

    const int pbase = w * 128;
    for (int ps = 0; ps < 32; ++ps) {
        const int p = pbase + ps * 4 + 2 * lh;
        v2f af[4], bf[4];
#pragma unroll
        for (int mi = 0; mi < 4; ++mi) {          // A[m][p'] = K[p'][m]
            int m = mi * 16 + lr;
            af[mi].x = Kp[(size_t)p * DH_ + m];
            af[mi].y = Kp[(size_t)(p + 1) * DH_ + m];
        }
#pragma unroll
        for (int ni = 0; ni < 4; ++ni) {          // B[p'][n] = V[p'][n]
            int n = ni * 16 + lr;
            bf[ni].x = Vp[(size_t)p * DH_ + n];
            bf[ni].y = Vp[(size_t)(p + 1) * DH_ + n];
        }
#pragma unroll
        for (int mi = 0; mi < 4; ++mi)
#pragma unroll
            for (int ni = 0; ni < 4; ++ni)
                acc[mi][ni] = wmma4(af[mi], bf[ni], acc[mi][ni]);
    }

    __shared__ float red[8][16][17];
    float* Mo = Mb + (size_t)bh * (DH_ * DH_);
    for (int mi = 0; mi < 4; ++mi) {
        for (int ni = 0; ni < 4; ++ni) {
#pragma unroll
            for (int v = 0; v < 8; ++v) red[w][v + 8 * lh][lr] = acc[mi][ni][v];
            __syncthreads();
            {
                int rr = t >> 4, cc = t & 15;     // 256 threads = 16x16 tile
                float s = 0.f;
#pragma unroll
                for (int ww = 0; ww < 8; ++ww) s += red[ww][rr][cc];
                Mo[(mi * 16 + rr) * DH_ + (ni * 16 + cc)] = s;
            }
            __syncthreads();
        }
    }
}

// ---------------------------------------------------------------------------
// Kernel 4: O = Q @ M per (b,h): (1024x64)@(64x64), written into (B,P,HID).
// One wave per 64-row tile; M is tiny and stays hot in cache.
// ---------------------------------------------------------------------------
__global__ __launch_bounds__(32) void qm_kernel(const float* __restrict__ qb,
                                                const float* __restrict__ Mb,
                                                float* __restrict__ Ob) {
    const int bh = blockIdx.x;
    const int m0 = blockIdx.y * 64;
    const int t  = threadIdx.x;
    const int lr = t & 15, lh = (t >> 4) & 1;
    const float* Qp = qb + (size_t)bh * P_ * DH_;
    const float* Mh = Mb + (size_t)bh * (DH_ * DH_);

    v8f acc[4][4];
#pragma unroll
    for (int mi = 0; mi < 4; ++mi)
#pragma unroll
        for (int ni = 0; ni < 4; ++ni) acc[mi][ni] = v8f_zero();

    for (int ks = 0; ks < 16; ++ks) {
        const int kk = ks * 4 + 2 * lh;
        v2f af[4], bf[4];
#pragma unroll
        for (int mi = 0; mi < 4; ++mi) {
            int m = m0 + mi * 16 + lr;
            af[mi].x = Qp[(size_t)m * DH_ + kk];
            af[mi].y = Qp[(size_t)m * DH_ + kk + 1];
        }
#pragma unroll
        for (int ni = 0; ni < 4; ++ni) {
            int n = ni * 16 + lr;
            bf[ni].x = Mh[kk * DH_ + n];
            bf[ni].y = Mh[(kk + 1) * DH_ + n];
        }
#pragma unroll
        for (int mi = 0; mi < 4; ++mi)
#pragma unroll
            for (int ni = 0; ni < 4; ++ni)
                acc[mi][ni] = wmma4(af[mi], bf[ni], acc[mi][ni]);
    }

    const int b = bh / HEADS_, h = bh % HEADS_;
#pragma unroll
    for (int mi = 0; mi < 4; ++mi)
#pragma unroll
        for (int ni = 0; ni < 4; ++ni) {
            int col = h * DH_ + ni * 16 + lr;
#pragma unroll
            for (int v = 0; v < 8; ++v) {
                int p = m0 + mi * 16 + v + 8 * lh;
                Ob[((size_t)b * P_ + p) * HID_ + col] = acc[mi][ni][v];
            }
        }
}

// ---------------------------------------------------------------------------
// Kernel 5: out = relu(O @ lin_w + lin_b) + x   (16384x768 @ 768x768)
// Same pipelined 256x128 tiling as the QKV GEMM.
// ---------------------------------------------------------------------------
__global__ __launch_bounds__(256) void out_kernel(
    const float* __restrict__ Ob, const float* __restrict__ lw,
    const float* __restrict__ lbias, const float* __restrict__ x,
    float* __restrict__ out) {
    __shared__ float As[256][20];
    __shared__ float Bs[16][132];

    const int t    = threadIdx.x;
    const int lane = t & 31;
    const int lr   = lane & 15;
    const int lh   = lane >> 4;
    const int w    = t >> 5;
    const int wm   = w >> 1, wn = w & 1;
    const int n0   = blockIdx.x * 128;
    const int m0   = blockIdx.y * 256;

    v8f acc[4][4];
#pragma unroll
    for (int mi = 0; mi < 4; ++mi)
#pragma unroll
        for (int ni = 0; ni < 4; ++ni) acc[mi][ni] = v8f_zero();

    float4 aReg[4], bReg[2];
#pragma unroll
    for (int i = 0; i < 4; ++i) {
        int lin = i * 256 + t;
        int row = lin >> 2, c4 = (lin & 3) * 4;
        aReg[i] = *(const float4*)(Ob + (size_t)(m0 + row) * HID_ + c4);
    }
#pragma unroll
    for (int i = 0; i < 2; ++i) {
        int lin = i * 256 + t;
        int row = lin >> 5, c4 = (lin & 31) * 4;
        bReg[i] = *(const float4*)(lw + (size_t)row * E_ + n0 + c4);
    }

    for (int k0 = 0; k0 < HID_; k0 += 16) {
#pragma unroll
        for (int i = 0; i < 4; ++i) {
            int lin = i * 256 + t;
            int row = lin >> 2, c4 = (lin & 3) * 4;
            *(float4*)&As[row][c4] = aReg[i];
        }
#pragma unroll
        for (int i = 0; i < 2; ++i) {
            int lin = i * 256 + t;
            int row = lin >> 5, c4 = (lin & 31) * 4;
            *(float4*)&Bs[row][c4] = bReg[i];
        }
        __syncthreads();

        if (k0 + 16 < HID_) {
            const int kn = k0 + 16;
#pragma unroll
            for (int i = 0; i < 4; ++i) {
                int lin = i * 256 + t;
                int row = lin >> 2, c4 = (lin & 3) * 4;
                aReg[i] = *(const float4*)(Ob + (size_t)(m0 + row) * HID_ + kn + c4);
            }
#pragma unroll
            for (int i = 0; i < 2; ++i) {
                int lin = i * 256 + t;
                int row = lin >> 5, c4 = (lin & 31) * 4;
                bReg[i] = *(const float4*)(lw + (size_t)(kn + row) * E_ + n0 + c4);
            }
        }

#pragma unroll
        for (int ks = 0; ks < 4; ++ks) {
            const int kk = ks * 4 + 2 * lh;
            v2f af[4], bf[4];
#pragma unroll
            for (int mi = 0; mi < 4; ++mi) {
                int m = wm * 64 + mi * 16 + lr;
                af[mi].x = As[m][kk];
                af[mi].y = As[m][kk + 1];
            }
#pragma unroll
            for (int ni = 0; ni < 4; ++ni) {
                int n = wn * 64 + ni * 16 + lr;
                bf[ni].x = Bs[kk][n];
                bf[ni].y = Bs[kk + 1][n];
            }
#pragma unroll
            for (int mi = 0; mi < 4; ++mi)
#pragma unroll
                for (int ni = 0; ni < 4; ++ni)
                    acc[mi][ni] = wmma4(af[mi], bf[ni], acc[mi][ni]);
        }
        __syncthreads();
    }

#pragma unroll
    for (int mi = 0; mi < 4; ++mi)
#pragma unroll
        for (int ni = 0; ni < 4; ++ni) {
            int n = n0 + wn * 64 + ni * 16 + lr;
            float bia = lbias[n];
#pragma unroll
            for (int v = 0; v < 8; ++v) {
                int gm = m0 + wm * 64 + mi * 16 + v + 8 * lh;
                float val = acc[mi][ni][v] + bia;
                val = fmaxf(val, 0.f);
                out[(size_t)gm * E_ + n] = val + x[(size_t)gm * E_ + n];
            }
        }
}

// ---------------------------------------------------------------------------
extern "C" void kernel_launch(void* const* d_in, const int* in_sizes, int n_in,
                              void* d_out, int out_size, void* d_ws, size_t ws_size,
                              hipStream_t stream) {
    (void)in_sizes; (void)n_in; (void)out_size; (void)ws_size;
    const float* x   = (const float*)d_in[0];
    const float* lnw = (const float*)d_in[1];
    const float* lnb = (const float*)d_in[2];
    const float* qkv = (const float*)d_in[3];
    const float* lw  = (const float*)d_in[4];
    const float* lb  = (const float*)d_in[5];
    float* out = (float*)d_out;

    float* ws    = (float*)d_ws;
    float* stats = ws;                                        // 32 floats
    float* qb    = ws + 64;
    float* kb    = qb + (size_t)B_ * HEADS_ * P_ * DH_;       // 12.58M each
    float* vb    = kb + (size_t)B_ * HEADS_ * P_ * DH_;
    float* Mb    = vb + (size_t)B_ * HEADS_ * P_ * DH_;       // 786432
    float* Ob    = Mb + (size_t)B_ * HEADS_ * DH_ * DH_;      // 12.58M

    stats_kernel<<<B_, 256, 0, stream>>>(x, stats);
    qkv_kernel<<<dim3(QKV_N / 128, (B_ * P_) / 256), 256, 0, stream>>>(
        x, lnw, lnb, qkv, stats, qb, kb, vb);
    ktv_kernel<<<B_ * HEADS_, 256, 0, stream>>>(kb, vb, Mb);
    qm_kernel<<<dim3(B_ * HEADS_, P_ / 64), 32, 0, stream>>>(qb, Mb, Ob);
    out_kernel<<<dim3(E_ / 128, (B_ * P_) / 256), 256, 0, stream>>>(
        Ob, lw, lb, x, out);
}